// SlotGCN_32916629357419
// MI455X (gfx1250) — compile-verified
//
#include <hip/hip_runtime.h>
#include <hip/hip_bf16.h>

typedef __attribute__((ext_vector_type(2))) float v2f;
typedef __attribute__((ext_vector_type(8))) float v8f;

static __device__ __forceinline__ void atomic_fadd(float* p, float v) {
    __hip_atomic_fetch_add(p, v, __ATOMIC_RELAXED, __HIP_MEMORY_SCOPE_AGENT);
}

// ---------------------------------------------------------------------------
// Degree / normalization helpers
// ---------------------------------------------------------------------------
__global__ void gcn_init_deg(float* __restrict__ deg, int n) {
    int i = blockIdx.x * blockDim.x + threadIdx.x;
    if (i < n) deg[i] = 1.0f;                     // self-loop
}

__global__ void gcn_count_deg(const int* __restrict__ dst, int E,
                              float* __restrict__ deg) {
    int e = blockIdx.x * blockDim.x + threadIdx.x;
    if (e < E) atomic_fadd(&deg[dst[e]], 1.0f);
}

__global__ void gcn_rsqrt(float* __restrict__ deg, int n) {
    int i = blockIdx.x * blockDim.x + threadIdx.x;
    if (i < n) deg[i] = rsqrtf(deg[i]);
}

__global__ void gcn_zero(float* __restrict__ p, long long n) {
    long long i = (long long)blockIdx.x * blockDim.x + threadIdx.x;
    if (i < n) p[i] = 0.0f;
}

// ---------------------------------------------------------------------------
// fp32 WMMA GEMM:  C[M,N] = A[M,K] @ B[K,N]   (row-major, M%16==0, N%64==0,
// K%4==0).  One wave computes a 16x64 strip: 4 accumulators share one A
// fragment per k-step -> 4 V_WMMA_F32_16X16X4_F32 per A load.
//   blockDim.x = 128 (4 waves); wave w -> mTile = blockIdx.x*4 + w
//   grid.x = ceil(M/64), grid.y = N/64
// ---------------------------------------------------------------------------
__global__ void gcn_gemm_f32_wmma(const float* __restrict__ A,
                                  const float* __restrict__ B,
                                  float* __restrict__ C,
                                  int M, int N, int K) {
    const int wave  = threadIdx.x >> 5;
    const int lane  = threadIdx.x & 31;
    const int mTile = blockIdx.x * 4 + wave;
    if (mTile * 16 >= M) return;                  // wave-uniform: EXEC stays all-1

    const int r16   = lane & 15;                  // A row / B col within tile
    const int koff  = (lane >> 4) << 1;           // 0 for lanes 0-15, 2 for 16-31
    const int nBase = blockIdx.y * 64 + r16;      // this lane's column in 4 n-tiles

    v8f c0 = {0.f,0.f,0.f,0.f,0.f,0.f,0.f,0.f};
    v8f c1 = c0, c2 = c0, c3 = c0;

    const float* Arow = A + ((long long)mTile * 16 + r16) * K;

    for (int k0 = 0; k0 < K; k0 += 4) {
        // A 16x4 fragment: VGPR0 = A[m][kbase], VGPR1 = A[m][kbase+1]
        v2f a = *(const v2f*)(Arow + k0 + koff);
        // B 4x16 fragments for 4 adjacent n-tiles
        const float* Brow0 = B + (long long)(k0 + koff)     * N + nBase;
        const float* Brow1 = B + (long long)(k0 + koff + 1) * N + nBase;
        v2f b0, b1, b2, b3;
        b0.x = Brow0[0];  b0.y = Brow1[0];
        b1.x = Brow0[16]; b1.y = Brow1[16];
        b2.x = Brow0[32]; b2.y = Brow1[32];
        b3.x = Brow0[48]; b3.y = Brow1[48];
        c0 = __builtin_amdgcn_wmma_f32_16x16x4_f32(false, a, false, b0, (short)0, c0, false, false);
        c1 = __builtin_amdgcn_wmma_f32_16x16x4_f32(false, a, false, b1, (short)0, c1, false, false);
        c2 = __builtin_amdgcn_wmma_f32_16x16x4_f32(false, a, false, b2, (short)0, c2, false, false);
        c3 = __builtin_amdgcn_wmma_f32_16x16x4_f32(false, a, false, b3, (short)0, c3, false, false);
    }

    // C/D 16x16 f32 layout: VGPR v -> row v (lanes 0-15) / row v+8 (lanes 16-31)
    const long long rowBase = (long long)mTile * 16 + ((lane >> 4) << 3);
#pragma unroll
    for (int v = 0; v < 8; ++v) {
        float* Crow = C + (rowBase + v) * N + nBase;
        Crow[0]  = c0[v];
        Crow[16] = c1[v];
        Crow[32] = c2[v];
        Crow[48] = c3[v];
    }
}

// ---------------------------------------------------------------------------
// Edge scatter:  agg[dst] += h[src] * dinv[src] * dinv[dst]
// One edge per wave, lanes stride over C channels -> coalesced loads/atomics.
// ---------------------------------------------------------------------------
__global__ void gcn_scatter(const float* __restrict__ h,
                            const int* __restrict__ src,
                            const int* __restrict__ dst,
                            const float* __restrict__ dinv,
                            float* __restrict__ agg,
                            int E, int C) {
    const int wavesPerBlock = blockDim.x >> 5;
    const int e    = blockIdx.x * wavesPerBlock + (threadIdx.x >> 5);
    const int lane = threadIdx.x & 31;
    if (e >= E) return;
    const int s = src[e];
    const int d = dst[e];
    const float nrm = dinv[s] * dinv[d];
    const float* hs = h   + (long long)s * C;
    float*       ad = agg + (long long)d * C;
    for (int c = lane; c < C; c += 32) {
        atomic_fadd(&ad[c], hs[c] * nrm);
    }
}

// ---------------------------------------------------------------------------
// Epilogue: agg[i,c] += h[i,c] * dinv[i]^2 + b[c] ; optional ReLU (in place)
// ---------------------------------------------------------------------------
__global__ void gcn_finalize(const float* __restrict__ h,
                             const float* __restrict__ dinv,
                             const float* __restrict__ bias,
                             float* __restrict__ agg,
                             int nNodes, int C, int doRelu) {
    long long i = (long long)blockIdx.x * blockDim.x + threadIdx.x;
    long long total = (long long)nNodes * C;
    if (i >= total) return;
    int node = (int)(i / C);
    int c    = (int)(i % C);
    float di = dinv[node];
    float v  = agg[i] + h[i] * di * di + bias[c];
    if (doRelu) v = fmaxf(v, 0.0f);
    agg[i] = v;
}

// ---------------------------------------------------------------------------
// Launcher
// ---------------------------------------------------------------------------
extern "C" void kernel_launch(void* const* d_in, const int* in_sizes, int n_in,
                              void* d_out, int out_size, void* d_ws, size_t ws_size,
                              hipStream_t stream) {
    const float* x   = (const float*)d_in[0];
    const int*   ei  = (const int*)d_in[1];     // edge_index [2, E] (int32)
    const float* W1  = (const float*)d_in[2];
    const float* b1  = (const float*)d_in[3];
    const float* W2  = (const float*)d_in[4];
    const float* b2  = (const float*)d_in[5];

    const int IN_C  = 128;
    const int HID_C = 128;
    const int OUT_C = 64;
    const int nNodes = in_sizes[0] / IN_C;       // 50000
    const int E      = in_sizes[1] / 2;          // 600000
    const int* srcIdx = ei;
    const int* dstIdx = ei + E;

    // Workspace layout: dinv | bufA (N*128) | bufB (N*128)
    float* dinv = (float*)d_ws;
    float* bufA = dinv + nNodes;                 // H1, later H2 (first N*64)
    float* bufB = bufA + (long long)nNodes * HID_C;  // agg1, then Hrelu
    float* out  = (float*)d_out;                 // doubles as agg2

    const int T = 256;
    const long long nh = (long long)nNodes * HID_C;
    const long long no = (long long)nNodes * OUT_C;
    const int mBlocks = (nNodes / 16 + 3) / 4;   // 4 m-tiles (waves) per block

    // 1) degree (with self-loop) -> dinv = rsqrt(deg)
    gcn_init_deg <<<(nNodes + T - 1) / T, T, 0, stream>>>(dinv, nNodes);
    gcn_count_deg<<<(E + T - 1) / T, T, 0, stream>>>(dstIdx, E, dinv);
    gcn_rsqrt    <<<(nNodes + T - 1) / T, T, 0, stream>>>(dinv, nNodes);

    // 2) H1 = X @ W1   (M=nNodes, N=128, K=128)
    {
        dim3 grid(mBlocks, HID_C / 64);
        gcn_gemm_f32_wmma<<<grid, 128, 0, stream>>>(x, W1, bufA,
                                                    nNodes, HID_C, IN_C);
    }

    // 3) agg1 = scatter(H1)
    gcn_zero<<<(unsigned)((nh + T - 1) / T), T, 0, stream>>>(bufB, nh);
    gcn_scatter<<<(E + 7) / 8, 256, 0, stream>>>(bufA, srcIdx, dstIdx, dinv,
                                                 bufB, E, HID_C);

    // 4) Hrelu = relu(agg1 + H1*dinv^2 + b1)  (in place in bufB)
    gcn_finalize<<<(unsigned)((nh + T - 1) / T), T, 0, stream>>>(
        bufA, dinv, b1, bufB, nNodes, HID_C, /*relu=*/1);

    // 5) H2 = Hrelu @ W2  (M=nNodes, N=64, K=128) -> bufA (H1 is dead)
    {
        dim3 grid(mBlocks, OUT_C / 64);
        gcn_gemm_f32_wmma<<<grid, 128, 0, stream>>>(bufB, W2, bufA,
                                                    nNodes, OUT_C, HID_C);
    }

    // 6) agg2 = scatter(H2) directly into d_out
    gcn_zero<<<(unsigned)((no + T - 1) / T), T, 0, stream>>>(out, no);
    gcn_scatter<<<(E + 7) / 8, 256, 0, stream>>>(bufA, srcIdx, dstIdx, dinv,
                                                 out, E, OUT_C);

    // 7) out = agg2 + H2*dinv^2 + b2
    gcn_finalize<<<(unsigned)((no + T - 1) / T), T, 0, stream>>>(
        bufA, dinv, b2, out, nNodes, OUT_C, /*relu=*/0);
}